// PartitionedGraphMixer_85341000172244
// MI455X (gfx1250) — compile-verified
//
#include <hip/hip_runtime.h>
#include <hip/hip_bf16.h>
#include <math.h>

// ---------------- problem constants (from reference setup) ----------------
#define KN    64     // neighbors per node
#define FEAT  172
#define TDIM  100
#define FIN   272    // FEAT + TDIM
#define CDIM  64
#define TOKH  32
#define CHH   256
#define NLAYER 2
#define GDIM  100
#define NOUT  172
#define KOUT  236    // CDIM + FEAT
#define LN_EPS 1e-5f

// LDS strides (padded to break bank patterns; row strides stay 16B-aligned)
#define LDBIG 276
#define LDCH  260
#define LDX   68
#define LDHID 36

typedef float v2f __attribute__((ext_vector_type(2)));
typedef float v8f __attribute__((ext_vector_type(8)));
typedef int   v4i __attribute__((ext_vector_type(4)));

#define AS1 __attribute__((address_space(1)))
#define AS3 __attribute__((address_space(3)))

#if __has_builtin(__builtin_amdgcn_global_load_async_to_lds_b128) && \
    __has_builtin(__builtin_amdgcn_s_wait_asynccnt)
#define HAVE_ASYNC_LDS 1
#else
#define HAVE_ASYNC_LDS 0
#endif

#if HAVE_ASYNC_LDS
// one 16B chunk: global -> LDS, ASYNCcnt-tracked, no VGPR round trip
__device__ __forceinline__ void async_copy_b128(const float* gsrc, float* ldst) {
  __builtin_amdgcn_global_load_async_to_lds_b128(
      (AS1 v4i*)(uintptr_t)gsrc,
      (AS3 v4i*)(uint32_t)(uintptr_t)ldst, 0, 0);
}
#endif

// D = A(16 x 4*ksteps, row-major) * B(4*ksteps x 16, row-major) + acc
// via V_WMMA_F32_16X16X4_F32.  Per-lane operand layout (ISA 7.12.2):
//   A: lanes 0-15 -> M=lane, K={0,1}; lanes 16-31 -> M=lane-16, K={2,3}
//   B: lanes 0-15 -> K={0,1}, N=lane; lanes 16-31 -> K={2,3}, N=lane-16
//   C/D: v8f, row = (lane>=16)*8 + i, col = lane&15
__device__ __forceinline__ v8f wmma_f32_kloop(const float* Arow, const float* Bcol,
                                              int ldb, int ksteps, v8f acc) {
  const int half = (threadIdx.x >> 4) & 1;
  for (int kk = 0; kk < ksteps; ++kk) {
    const int k0 = (kk << 2) + (half << 1);
    v2f a = { Arow[k0], Arow[k0 + 1] };
    v2f b = { Bcol[(size_t)k0 * ldb], Bcol[(size_t)(k0 + 1) * ldb] };
    acc = __builtin_amdgcn_wmma_f32_16x16x4_f32(false, a, false, b, (short)0, acc,
                                                false, false);
  }
  return acc;
}

__device__ __forceinline__ v8f wmma_f32_tile(const float* Abase, int lda,
                                             const float* Bbase, int ldb,
                                             int ksteps, v8f acc) {
  const int l16 = threadIdx.x & 15;
  return wmma_f32_kloop(Abase + (size_t)l16 * lda, Bbase + l16, ldb, ksteps, acc);
}

__device__ __forceinline__ void store_tile(float* D, int ldd, v8f c) {
  const int lane = threadIdx.x & 31;
  const int half = lane >> 4, l16 = lane & 15;
#pragma unroll
  for (int i = 0; i < 8; ++i) D[(size_t)(half * 8 + i) * ldd + l16] = c[i];
}

__device__ __forceinline__ float gelu_exact(float x) {
  return 0.5f * x * (1.0f + erff(x * 0.70710678118654752440f));
}

// =====================================================================
// Kernel A: per-b gather + time-encode + projection + 2 mixer layers + agg
// one workgroup (256 threads, 8 waves) per batch row b
// =====================================================================
__global__ __launch_bounds__(256) void mixer_kernel(
    const float* __restrict__ edge_table, const float* __restrict__ time_w,
    const float* __restrict__ proj_W, const float* __restrict__ proj_b,
    const float* __restrict__ tok_ln_s, const float* __restrict__ tok_ln_b,
    const float* __restrict__ tok_W1, const float* __restrict__ tok_b1,
    const float* __restrict__ tok_W2, const float* __restrict__ tok_b2,
    const float* __restrict__ ch_ln_s, const float* __restrict__ ch_ln_b,
    const float* __restrict__ ch_W1, const float* __restrict__ ch_b1,
    const float* __restrict__ ch_W2, const float* __restrict__ ch_b2,
    const int* __restrict__ nbr_nid, const int* __restrict__ nbr_eid,
    const float* __restrict__ t0s, const float* __restrict__ nbr_t,
    float* __restrict__ aggOut) {
  __shared__ float sBig[KN * LDBIG];   // 64x272 staged input, later 64x256 ch-hidden
  __shared__ float sX[KN * LDX];       // running x (tokens x channels)
  __shared__ float sLn[KN * LDX];      // layer-normed operand
  __shared__ float sHid[KN * LDHID];   // token hidden 64x32
  __shared__ float sRed[512];
  __shared__ float sMean[64];
  __shared__ float sRstd[64];
  __shared__ int   sEid[KN];
  __shared__ int   sNid[KN];
  __shared__ float sDt[KN];

  const int b = blockIdx.x;
  const int tid = threadIdx.x;
  const int wave = tid >> 5;
  const int lane = tid & 31;
  const int half = lane >> 4, l16 = lane & 15;

  // ---- stage neighbor ids / time deltas once ----
  if (tid < KN) {
    sEid[tid] = nbr_eid[(size_t)b * KN + tid];
    sNid[tid] = nbr_nid[(size_t)b * KN + tid];
    sDt[tid]  = t0s[b] - nbr_t[(size_t)b * KN + tid];
  }
  __syncthreads();

  // ---- stage concat(edge_feats, time_feats) : 64 x 272 ----
  // edge part: 64 rows x 43 aligned 16B chunks (172 f32 = 43 float4 exactly)
#if HAVE_ASYNC_LDS
  for (int idx = tid; idx < KN * 43; idx += 256) {
    const int r = idx / 43, q = idx - r * 43;
    async_copy_b128(edge_table + (size_t)sEid[r] * FEAT + q * 4,
                    &sBig[(size_t)r * LDBIG + q * 4]);
  }
#else
  for (int idx = tid; idx < KN * 43; idx += 256) {
    const int r = idx / 43, q = idx - r * 43;
    const float4 v = *(const float4*)(edge_table + (size_t)sEid[r] * FEAT + q * 4);
    *(float4*)(&sBig[(size_t)r * LDBIG + q * 4]) = v;
  }
#endif
  // time-encoding part (overlaps with the async DMA above)
  for (int idx = tid; idx < KN * TDIM; idx += 256) {
    const int r = idx / TDIM, j = idx - r * TDIM;
    sBig[(size_t)r * LDBIG + FEAT + j] =
        (sNid[r] == 0) ? 0.0f : cosf(sDt[r] * time_w[j]);
  }
#if HAVE_ASYNC_LDS
  __builtin_amdgcn_s_wait_asynccnt(0);
#endif
  __syncthreads();

  // ---- projection GEMM: x(64x64) = in(64x272) @ proj_W(272x64) + b ----
  for (int t = wave; t < 16; t += 8) {
    const int mi = t >> 2, ni = t & 3;
    v8f c = {};
    c = wmma_f32_tile(&sBig[(size_t)mi * 16 * LDBIG], LDBIG,
                      &proj_W[ni * 16], CDIM, FIN / 4, c);
    const float bias = proj_b[ni * 16 + l16];
#pragma unroll
    for (int i = 0; i < 8; ++i) c[i] += bias;
    store_tile(&sX[(size_t)(mi * 16) * LDX + ni * 16], LDX, c);
  }
  __syncthreads();

  for (int l = 0; l < NLAYER; ++l) {
    // ================= token mixing =================
    {  // LN over K (per channel), result stored transposed: sLn[c][k]
      const int c = tid & 63, q = tid >> 6;
      float s1 = 0.f, s2 = 0.f;
      for (int k = q * 16; k < q * 16 + 16; ++k) {
        const float v = sX[(size_t)k * LDX + c];
        s1 += v; s2 += v * v;
      }
      sRed[q * 64 + c] = s1; sRed[256 + q * 64 + c] = s2;
      __syncthreads();
      if (tid < 64) {
        float m = 0.f, ss = 0.f;
        for (int q2 = 0; q2 < 4; ++q2) { m += sRed[q2 * 64 + tid]; ss += sRed[256 + q2 * 64 + tid]; }
        m *= (1.0f / 64.0f);
        ss = ss * (1.0f / 64.0f) - m * m;
        sMean[tid] = m; sRstd[tid] = rsqrtf(ss + LN_EPS);
      }
      __syncthreads();
      for (int idx = tid; idx < 64 * 64; idx += 256) {
        const int cc = idx >> 6, k = idx & 63;
        sLn[(size_t)cc * LDX + k] =
            (sX[(size_t)k * LDX + cc] - sMean[cc]) * sRstd[cc]
            * tok_ln_s[l * KN + k] + tok_ln_b[l * KN + k];
      }
      __syncthreads();
    }
    // hid(64x32) = gelu(sLn(64x64) @ tok_W1(64x32) + b1)
    for (int t = wave; t < 8; t += 8) {
      const int mi = t >> 1, ni = t & 1;
      v8f c = {};
      c = wmma_f32_tile(&sLn[(size_t)mi * 16 * LDX], LDX,
                        &tok_W1[(size_t)l * KN * TOKH + ni * 16], TOKH, 16, c);
      const float bias = tok_b1[l * TOKH + ni * 16 + l16];
#pragma unroll
      for (int i = 0; i < 8; ++i) c[i] = gelu_exact(c[i] + bias);
      store_tile(&sHid[(size_t)(mi * 16) * LDHID + ni * 16], LDHID, c);
    }
    __syncthreads();
    // out(64x64) = hid @ tok_W2(32x64) + b2 ; x[k][c] += out[c][k]
    for (int t = wave; t < 16; t += 8) {
      const int mi = t >> 2, ni = t & 3;
      v8f c = {};
      c = wmma_f32_tile(&sHid[(size_t)mi * 16 * LDHID], LDHID,
                        &tok_W2[(size_t)l * TOKH * KN + ni * 16], KN, 8, c);
      const int col = ni * 16 + l16;                 // token index k
      const float bias = tok_b2[l * KN + col];
#pragma unroll
      for (int i = 0; i < 8; ++i) {
        const int row = mi * 16 + half * 8 + i;      // channel index c
        sX[(size_t)col * LDX + row] += c[i] + bias;  // transposed residual
      }
    }
    __syncthreads();

    // ================= channel mixing =================
    {  // LN over C (per token row)
      const int k = tid & 63, q = tid >> 6;
      float s1 = 0.f, s2 = 0.f;
      for (int c = q * 16; c < q * 16 + 16; ++c) {
        const float v = sX[(size_t)k * LDX + c];
        s1 += v; s2 += v * v;
      }
      sRed[q * 64 + k] = s1; sRed[256 + q * 64 + k] = s2;
      __syncthreads();
      if (tid < 64) {
        float m = 0.f, ss = 0.f;
        for (int q2 = 0; q2 < 4; ++q2) { m += sRed[q2 * 64 + tid]; ss += sRed[256 + q2 * 64 + tid]; }
        m *= (1.0f / 64.0f);
        ss = ss * (1.0f / 64.0f) - m * m;
        sMean[tid] = m; sRstd[tid] = rsqrtf(ss + LN_EPS);
      }
      __syncthreads();
      for (int idx = tid; idx < 64 * 64; idx += 256) {
        const int k2 = idx >> 6, c = idx & 63;
        sLn[(size_t)k2 * LDX + c] =
            (sX[(size_t)k2 * LDX + c] - sMean[k2]) * sRstd[k2]
            * ch_ln_s[l * CDIM + c] + ch_ln_b[l * CDIM + c];
      }
      __syncthreads();
    }
    // big(64x256) = gelu(sLn(64x64) @ ch_W1(64x256) + b1)
    for (int t = wave; t < 64; t += 8) {
      const int mi = t >> 4, ni = t & 15;
      v8f c = {};
      c = wmma_f32_tile(&sLn[(size_t)mi * 16 * LDX], LDX,
                        &ch_W1[(size_t)l * CDIM * CHH + ni * 16], CHH, 16, c);
      const float bias = ch_b1[l * CHH + ni * 16 + l16];
#pragma unroll
      for (int i = 0; i < 8; ++i) c[i] = gelu_exact(c[i] + bias);
      store_tile(&sBig[(size_t)(mi * 16) * LDCH + ni * 16], LDCH, c);
    }
    __syncthreads();
    // x(64x64) += big(64x256) @ ch_W2(256x64) + b2
    for (int t = wave; t < 16; t += 8) {
      const int mi = t >> 2, ni = t & 3;
      v8f c = {};
      c = wmma_f32_tile(&sBig[(size_t)mi * 16 * LDCH], LDCH,
                        &ch_W2[(size_t)l * CHH * CDIM + ni * 16], CDIM, CHH / 4, c);
      const float bias = ch_b2[l * CDIM + ni * 16 + l16];
#pragma unroll
      for (int i = 0; i < 8; ++i) {
        const int row = mi * 16 + half * 8 + i;      // token index k
        sX[(size_t)row * LDX + ni * 16 + l16] += c[i] + bias;
      }
    }
    __syncthreads();
  }

  // ---- agg = mean over K ----
  {
    const int c = tid & 63, q = tid >> 6;
    float s = 0.f;
    for (int k = q * 16; k < q * 16 + 16; ++k) s += sX[(size_t)k * LDX + c];
    sRed[q * 64 + c] = s;
    __syncthreads();
    if (tid < 64) {
      const float m = sRed[tid] + sRed[64 + tid] + sRed[128 + tid] + sRed[192 + tid];
      aggOut[(size_t)b * CDIM + tid] = m * (1.0f / 64.0f);
    }
  }
}

// =====================================================================
// Kernel B: softmax(validity)-weighted neighbor mean + self feature
// softmax over {1, -1e10} == uniform 1/n_valid over valid ids
// (all-invalid corner: softmax is uniform 1/G over everything)
// =====================================================================
__global__ __launch_bounds__(192) void gap_kernel(
    const float* __restrict__ node_table, const int* __restrict__ node_ids,
    const int* __restrict__ gap_ids, float* __restrict__ h2) {
  __shared__ int sIds[GDIM];
  __shared__ int sNv;
  const int b = blockIdx.x;
  const int tid = threadIdx.x;
  if (tid < GDIM) sIds[tid] = gap_ids[(size_t)b * GDIM + tid];
  __syncthreads();
  if (tid == 0) {
    int nv = 0;
    for (int g = 0; g < GDIM; ++g) nv += (sIds[g] > 0);
    sNv = nv;
  }
  __syncthreads();
  if (tid < FEAT) {
    const int nv = sNv;
    float acc = 0.f;
    if (nv > 0) {
      for (int g = 0; g < GDIM; ++g) {
        const int id = sIds[g];
        if (id > 0) acc += node_table[(size_t)id * FEAT + tid];
      }
      acc *= 1.0f / ((float)nv * (float)GDIM);
    } else {
      for (int g = 0; g < GDIM; ++g) acc += node_table[(size_t)sIds[g] * FEAT + tid];
      acc *= 1.0f / ((float)GDIM * (float)GDIM);
    }
    const int nid = node_ids[b];
    h2[(size_t)b * FEAT + tid] = acc + node_table[(size_t)nid * FEAT + tid];
  }
}

// =====================================================================
// Kernel C: out(B x 172) = [agg | h2](B x 236) @ out_W(236 x 172) + out_b
// 16-row tile per block; 11 waves -> 11 column tiles (last clamped)
// =====================================================================
__global__ __launch_bounds__(352) void out_gemm_kernel(
    const float* __restrict__ agg, const float* __restrict__ h2,
    const float* __restrict__ out_W, const float* __restrict__ out_b,
    float* __restrict__ out) {
  __shared__ float sA[16 * 240];
  const int b0 = blockIdx.x * 16;
  const int tid = threadIdx.x;
  for (int idx = tid; idx < 16 * KOUT; idx += 352) {
    const int r = idx / KOUT, c = idx - r * KOUT;
    const float v = (c < CDIM) ? agg[(size_t)(b0 + r) * CDIM + c]
                               : h2[(size_t)(b0 + r) * FEAT + (c - CDIM)];
    sA[(size_t)r * 240 + c] = v;
  }
  __syncthreads();

  const int wave = tid >> 5;
  const int lane = tid & 31;
  const int half = lane >> 4, l16 = lane & 15;
  const int n0 = wave * 16;
  const int col = n0 + l16;
  const int colB = (col < NOUT) ? col : (NOUT - 1);   // clamp OOB B columns

  v8f c = {};
  c = wmma_f32_kloop(&sA[(size_t)l16 * 240], &out_W[colB], NOUT, KOUT / 4, c);

  if (col < NOUT) {
    const float bias = out_b[col];
#pragma unroll
    for (int i = 0; i < 8; ++i)
      out[(size_t)(b0 + half * 8 + i) * NOUT + col] = c[i] + bias;
  }
}

// =====================================================================
extern "C" void kernel_launch(void* const* d_in, const int* in_sizes, int n_in,
                              void* d_out, int out_size, void* d_ws, size_t ws_size,
                              hipStream_t stream) {
  const float* node_table = (const float*)d_in[0];
  const float* edge_table = (const float*)d_in[1];
  const float* time_w     = (const float*)d_in[2];
  const float* proj_W     = (const float*)d_in[3];
  const float* proj_b     = (const float*)d_in[4];
  const float* tok_ln_s   = (const float*)d_in[5];
  const float* tok_ln_b   = (const float*)d_in[6];
  const float* tok_W1     = (const float*)d_in[7];
  const float* tok_b1     = (const float*)d_in[8];
  const float* tok_W2     = (const float*)d_in[9];
  const float* tok_b2     = (const float*)d_in[10];
  const float* ch_ln_s    = (const float*)d_in[11];
  const float* ch_ln_b    = (const float*)d_in[12];
  const float* ch_W1      = (const float*)d_in[13];
  const float* ch_b1      = (const float*)d_in[14];
  const float* ch_W2      = (const float*)d_in[15];
  const float* ch_b2      = (const float*)d_in[16];
  const float* out_W      = (const float*)d_in[17];
  const float* out_b      = (const float*)d_in[18];
  const int*   node_ids   = (const int*)d_in[19];
  const int*   nbr_nid    = (const int*)d_in[20];
  const int*   nbr_eid    = (const int*)d_in[21];
  const int*   gap_ids    = (const int*)d_in[22];
  const float* t0s        = (const float*)d_in[23];
  const float* nbr_t      = (const float*)d_in[24];

  const int B = in_sizes[19];            // 8192
  float* out = (float*)d_out;
  float* aggWS = (float*)d_ws;                       // B x 64
  float* h2WS  = aggWS + (size_t)B * CDIM;           // B x 172

  mixer_kernel<<<B, 256, 0, stream>>>(
      edge_table, time_w, proj_W, proj_b,
      tok_ln_s, tok_ln_b, tok_W1, tok_b1, tok_W2, tok_b2,
      ch_ln_s, ch_ln_b, ch_W1, ch_b1, ch_W2, ch_b2,
      nbr_nid, nbr_eid, t0s, nbr_t, aggWS);

  gap_kernel<<<B, 192, 0, stream>>>(node_table, node_ids, gap_ids, h2WS);

  out_gemm_kernel<<<B / 16, 352, 0, stream>>>(aggWS, h2WS, out_W, out_b, out);
}